// SSM_59210419143299
// MI455X (gfx1250) — compile-verified
//
#include <hip/hip_runtime.h>
#include <stdint.h>

// ---------------------------------------------------------------------------
// Mamba selective scan (B=2, D=1536, L=2048, N=16), fp32.
// 3-pass chunked parallel scan:
//   pass1: per (b,d,chunk): P = prod(dA), S = h_end given h_in = 0
//   pass2: per (b,d,n): prefix-combine the 16 chunks -> Hin per chunk
//   pass3: per (b,d,chunk): re-run chunk from Hin, emit (y + x*D)*silu(z)
// B/C chunk tiles staged into LDS via gfx1250 async global->LDS loads
// (ASYNCcnt path), tiles read back as ds_load_b128.
// ---------------------------------------------------------------------------

namespace {
constexpr int kB        = 2;
constexpr int kD        = 1536;
constexpr int kL        = 2048;
constexpr int kN        = 16;
constexpr int kChunks   = 16;
constexpr int kChunkLen = kL / kChunks;   // 128
constexpr int kTPB      = 256;            // 8 wave32 per block
constexpr int kDBlks    = kD / kTPB;      // 6 blocks cover D
constexpr size_t kStateStride = (size_t)kB * kChunks * kD * kN;  // 786432 floats
constexpr float kLog2e = 1.44269504088896340736f;
constexpr float kLn2   = 0.69314718055994530942f;
}

#if defined(__HIP_DEVICE_COMPILE__) && \
    __has_builtin(__builtin_amdgcn_global_load_async_to_lds_b128) && \
    __has_builtin(__builtin_amdgcn_s_wait_asynccnt)
#define USE_ASYNC_LDS 1
#else
#define USE_ASYNC_LDS 0
#endif

#if USE_ASYNC_LDS
typedef int v4i_t __attribute__((ext_vector_type(4)));
typedef v4i_t __attribute__((address_space(1)))* gbl_v4i;   // global int4*
typedef v4i_t __attribute__((address_space(3)))* lds_v4i;   // LDS    int4*
#endif

__device__ __forceinline__ void tile_copy_b128(const float* g, float* l) {
#if USE_ASYNC_LDS
  __builtin_amdgcn_global_load_async_to_lds_b128(
      (gbl_v4i)(v4i_t*)const_cast<float*>(g),
      (lds_v4i)(v4i_t*)l,
      /*ioffset=*/0, /*cpol=*/0);
#else
  *(float4*)l = *(const float4*)g;
#endif
}

__device__ __forceinline__ void tile_wait() {
#if USE_ASYNC_LDS
  __builtin_amdgcn_s_wait_asynccnt(0);
#endif
}

// softplus(v) = ln(1 + e^v) = log2(1 + 2^(v*log2e)) * ln2, with overflow guard.
__device__ __forceinline__ float softplus_f(float v) {
  if (v > 20.0f) return v;
  const float e = __builtin_amdgcn_exp2f(v * kLog2e);
  return __builtin_amdgcn_logf(1.0f + e) * kLn2;
}

// silu(v) = v * sigmoid(v) = v * rcp(1 + 2^(-v*log2e))
__device__ __forceinline__ float silu_f(float v) {
  const float e = __builtin_amdgcn_exp2f(-v * kLog2e);
  return v * __builtin_amdgcn_rcpf(1.0f + e);
}

// Stage a [kN x kChunkLen] tile (2048 floats) of B or C into LDS.
__device__ __forceinline__ void stage_tile(const float* src_base, float* lds,
                                           int tid) {
#pragma unroll
  for (int i = 0; i < 2; ++i) {
    const int e   = (tid + i * kTPB) * 4;   // element index 0..2044
    const int row = e >> 7;                 // / kChunkLen
    const int col = e & (kChunkLen - 1);
    tile_copy_b128(src_base + (size_t)row * kL + col, &lds[e]);
  }
}

// ---------------------------------------------------------------------------
// Pass 1: per-chunk decay product P and partial state S (h_in = 0).
// ---------------------------------------------------------------------------
__global__ void __launch_bounds__(kTPB) ssm_pass1(
    const float* __restrict__ x, const float* __restrict__ delta,
    const float* __restrict__ A, const float* __restrict__ Bm,
    const float* __restrict__ delta_bias,
    float* __restrict__ P, float* __restrict__ S) {
  __shared__ float Bt[kN * kChunkLen];      // 8 KB

  const int blk  = blockIdx.x;
  const int dblk = blk % kDBlks;
  const int c    = (blk / kDBlks) % kChunks;
  const int b    = blk / (kDBlks * kChunks);
  const int tid  = threadIdx.x;
  const int d    = dblk * kTPB + tid;

  stage_tile(Bm + (size_t)b * kN * kL + c * kChunkLen, Bt, tid);
  tile_wait();
  __syncthreads();

  // A row, pre-scaled by log2(e) so dA = exp2(dl * Ad2[n])  (one v_exp each)
  float Ad2[kN];
  {
    const float4* a4 = (const float4*)(A + (size_t)d * kN);
#pragma unroll
    for (int i = 0; i < 4; ++i) {
      float4 v = a4[i];
      Ad2[4*i+0] = v.x * kLog2e; Ad2[4*i+1] = v.y * kLog2e;
      Ad2[4*i+2] = v.z * kLog2e; Ad2[4*i+3] = v.w * kLog2e;
    }
  }
  const float dbias = delta_bias[d];

  const size_t chanOff = ((size_t)b * kD + d) * kL + (size_t)c * kChunkLen;
  const float* dp = delta + chanOff;
  const float* xp = x + chanOff;

  float h[kN], p[kN];
#pragma unroll
  for (int n = 0; n < kN; ++n) { h[n] = 0.0f; p[n] = 1.0f; }

  for (int t0 = 0; t0 < kChunkLen; t0 += 4) {
    const float4 d4 = *(const float4*)(dp + t0);
    const float4 x4 = *(const float4*)(xp + t0);
    const float dv[4] = {d4.x, d4.y, d4.z, d4.w};
    const float xv[4] = {x4.x, x4.y, x4.z, x4.w};
    float dlv[4], uv[4];
#pragma unroll
    for (int j = 0; j < 4; ++j) {
      dlv[j] = softplus_f(dv[j] + dbias);
      uv[j]  = dlv[j] * xv[j];
    }
#pragma unroll
    for (int n = 0; n < kN; ++n) {
      const float4 b4 = *(const float4*)&Bt[n * kChunkLen + t0];  // ds_load_b128
      const float bb[4] = {b4.x, b4.y, b4.z, b4.w};
      float hn = h[n], pn = p[n];
#pragma unroll
      for (int j = 0; j < 4; ++j) {
        const float a = __builtin_amdgcn_exp2f(dlv[j] * Ad2[n]);
        pn *= a;
        hn = fmaf(a, hn, uv[j] * bb[j]);
      }
      h[n] = hn; p[n] = pn;
    }
  }

  const size_t o = (((size_t)b * kChunks + c) * kD + d) * kN;
#pragma unroll
  for (int i = 0; i < 4; ++i) {
    float4 pv = {p[4*i+0], p[4*i+1], p[4*i+2], p[4*i+3]};
    float4 hv = {h[4*i+0], h[4*i+1], h[4*i+2], h[4*i+3]};
    *(float4*)(P + o + 4*i) = pv;
    *(float4*)(S + o + 4*i) = hv;
  }
}

// ---------------------------------------------------------------------------
// Pass 2: prefix-combine chunk summaries -> incoming state per chunk.
// ---------------------------------------------------------------------------
__global__ void __launch_bounds__(kTPB) ssm_pass2(
    const float* __restrict__ P, const float* __restrict__ S,
    float* __restrict__ Hin) {
  const int tid = blockIdx.x * kTPB + threadIdx.x;  // 0 .. B*D*N-1
  const int n = tid & (kN - 1);
  const int d = (tid >> 4) % kD;
  const int b = tid / (kN * kD);
  float hs = 0.0f;
#pragma unroll
  for (int c = 0; c < kChunks; ++c) {
    const size_t idx = (((size_t)b * kChunks + c) * kD + d) * kN + n;
    Hin[idx] = hs;
    hs = fmaf(P[idx], hs, S[idx]);
  }
}

// ---------------------------------------------------------------------------
// Pass 3: re-run chunk with correct incoming state; emit gated output.
// ---------------------------------------------------------------------------
__global__ void __launch_bounds__(kTPB) ssm_pass3(
    const float* __restrict__ x, const float* __restrict__ delta,
    const float* __restrict__ A, const float* __restrict__ Bm,
    const float* __restrict__ Cm, const float* __restrict__ Dv,
    const float* __restrict__ z, const float* __restrict__ delta_bias,
    const float* __restrict__ Hin, float* __restrict__ out) {
  __shared__ float BCt[2 * kN * kChunkLen]; // 16 KB: B tile then C tile
  float* Bt = BCt;
  float* Ct = BCt + kN * kChunkLen;

  const int blk  = blockIdx.x;
  const int dblk = blk % kDBlks;
  const int c    = (blk / kDBlks) % kChunks;
  const int b    = blk / (kDBlks * kChunks);
  const int tid  = threadIdx.x;
  const int d    = dblk * kTPB + tid;

  stage_tile(Bm + (size_t)b * kN * kL + c * kChunkLen, Bt, tid);
  stage_tile(Cm + (size_t)b * kN * kL + c * kChunkLen, Ct, tid);
  tile_wait();
  __syncthreads();

  float Ad2[kN];
  {
    const float4* a4 = (const float4*)(A + (size_t)d * kN);
#pragma unroll
    for (int i = 0; i < 4; ++i) {
      float4 v = a4[i];
      Ad2[4*i+0] = v.x * kLog2e; Ad2[4*i+1] = v.y * kLog2e;
      Ad2[4*i+2] = v.z * kLog2e; Ad2[4*i+3] = v.w * kLog2e;
    }
  }
  const float dbias = delta_bias[d];
  const float Dd    = Dv[d];

  float h[kN];
  {
    const size_t o = (((size_t)b * kChunks + c) * kD + d) * kN;
#pragma unroll
    for (int i = 0; i < 4; ++i) {
      float4 v = *(const float4*)(Hin + o + 4*i);
      h[4*i+0] = v.x; h[4*i+1] = v.y; h[4*i+2] = v.z; h[4*i+3] = v.w;
    }
  }

  const size_t chanOff = ((size_t)b * kD + d) * kL + (size_t)c * kChunkLen;
  const float* dp = delta + chanOff;
  const float* xp = x + chanOff;
  const float* zp = z + chanOff;
  float*       op = out + chanOff;

  for (int t0 = 0; t0 < kChunkLen; t0 += 4) {
    const float4 d4 = *(const float4*)(dp + t0);
    const float4 x4 = *(const float4*)(xp + t0);
    const float4 z4 = *(const float4*)(zp + t0);
    const float dv[4] = {d4.x, d4.y, d4.z, d4.w};
    const float xv[4] = {x4.x, x4.y, x4.z, x4.w};
    const float zv[4] = {z4.x, z4.y, z4.z, z4.w};
    float dlv[4], uv[4];
    float yv[4] = {0.0f, 0.0f, 0.0f, 0.0f};
#pragma unroll
    for (int j = 0; j < 4; ++j) {
      dlv[j] = softplus_f(dv[j] + dbias);
      uv[j]  = dlv[j] * xv[j];
    }
#pragma unroll
    for (int n = 0; n < kN; ++n) {
      const float4 b4 = *(const float4*)&Bt[n * kChunkLen + t0];  // ds_load_b128
      const float4 c4 = *(const float4*)&Ct[n * kChunkLen + t0];  // ds_load_b128
      const float bb[4] = {b4.x, b4.y, b4.z, b4.w};
      const float cc[4] = {c4.x, c4.y, c4.z, c4.w};
      float hn = h[n];
#pragma unroll
      for (int j = 0; j < 4; ++j) {
        const float a = __builtin_amdgcn_exp2f(dlv[j] * Ad2[n]);
        hn = fmaf(a, hn, uv[j] * bb[j]);
        yv[j] = fmaf(hn, cc[j], yv[j]);
      }
      h[n] = hn;
    }
    float ov[4];
#pragma unroll
    for (int j = 0; j < 4; ++j) {
      ov[j] = (yv[j] + xv[j] * Dd) * silu_f(zv[j]);
    }
    float4 o4 = {ov[0], ov[1], ov[2], ov[3]};
    *(float4*)(op + t0) = o4;
  }
}

// ---------------------------------------------------------------------------
extern "C" void kernel_launch(void* const* d_in, const int* in_sizes, int n_in,
                              void* d_out, int out_size, void* d_ws,
                              size_t ws_size, hipStream_t stream) {
  (void)in_sizes; (void)n_in; (void)out_size; (void)ws_size;

  const float* x     = (const float*)d_in[0];
  const float* delta = (const float*)d_in[1];
  const float* A     = (const float*)d_in[2];
  const float* Bm    = (const float*)d_in[3];
  const float* Cm    = (const float*)d_in[4];
  const float* Dv    = (const float*)d_in[5];
  const float* z     = (const float*)d_in[6];
  const float* dbias = (const float*)d_in[7];
  float* out = (float*)d_out;

  float* P   = (float*)d_ws;                 // 3 MB
  float* S   = P + kStateStride;             // 3 MB
  float* Hin = S + kStateStride;             // 3 MB  (9.4 MB total in ws)

  const dim3 grid1(kB * kChunks * kDBlks);   // 192 blocks
  const dim3 grid2((kB * kD * kN) / kTPB);   // 192 blocks
  const dim3 block(kTPB);

  ssm_pass1<<<grid1, block, 0, stream>>>(x, delta, A, Bm, dbias, P, S);
  ssm_pass2<<<grid2, block, 0, stream>>>(P, S, Hin);
  ssm_pass3<<<grid1, block, 0, stream>>>(x, delta, A, Bm, Cm, Dv, z, dbias,
                                         Hin, out);
}